// DeepEquilibriumModel_46377056862864
// MI455X (gfx1250) — compile-verified
//
#include <hip/hip_runtime.h>
#include <hip/hip_bf16.h>
#include <stdint.h>

typedef float v2f __attribute__((ext_vector_type(2)));
typedef float v8f __attribute__((ext_vector_type(8)));

#define M_DIM 1024
#define N_DIM 2048
#define K_DIM 2048
#define BETA_C 0.9f
#define LAM_C 1e-4f

#define KC 32                 // K-chunk staged per double-buffer step
#define NC (K_DIM / KC)       // 64 chunks
#define A_STRIDE 36           // dwords; 144B rows: 16B-aligned, conflict-free frags
#define W_STRIDE 72           // dwords; 288B rows: 16B-aligned, conflict-free frags

// Branch-free tanh: tanh(x) = 1 - 2/(exp(2x)+1), via v_exp_f32 / v_rcp_f32.
__device__ __forceinline__ float fast_tanh(float x) {
  const float e = __builtin_amdgcn_exp2f(x * 2.885390081777927f); // 2*log2(e)
  return 1.0f - 2.0f * __builtin_amdgcn_rcpf(e + 1.0f);
}

__device__ __forceinline__ void async_ld_b128(uint32_t lds_byte_off,
                                              const float* gaddr) {
  asm volatile("global_load_async_to_lds_b128 %0, %1, off"
               :: "v"(lds_byte_off), "v"(gaddr) : "memory");
}

// ---------------------------------------------------------------------------
// GEMM via V_WMMA_F32_16X16X4_F32 (wave32) with double-buffered async
// global->LDS staging. Workgroup = 4 waves (128 thr), WG tile 128(M) x 64(N);
// each wave owns a 32x64 tile = 2x4 WMMA 16x16 tiles. A: MxK, W: KxN (row-maj).
// mode 0: out[m][n] = acc + addv[n]                       (x_proj = x@Wx + b)
// mode 1: out[m][n] = tanh(acc + addv[m*N+n]) - zin[..]   (g = f(z) - z)
// ---------------------------------------------------------------------------
__global__ __launch_bounds__(128) void gemm_wmma_f32(
    const float* __restrict__ A, const float* __restrict__ W,
    const float* __restrict__ addv, const float* __restrict__ zin,
    float* __restrict__ out, int mode)
{
  __shared__ float As[2][128 * A_STRIDE];   // 36864 B
  __shared__ float Ws[2][KC * W_STRIDE];    // 18432 B

  const int t    = threadIdx.x;
  const int lane = t & 31;
  const int wave = t >> 5;

  const int TN   = N_DIM / 64;              // 32 column tiles
  const int tm   = blockIdx.x / TN;
  const int tn   = blockIdx.x % TN;
  const int m0wg = tm * 128;
  const int n0   = tn * 64;
  const int m0   = m0wg + wave * 32;        // this wave's 32 rows

  const int row  = lane & 15;               // M (A frag) / N (B,C,D frag)
  const int kgrp = (lane >> 4) * 2;         // K pair selected by lane half

  const uint32_t as_base = (uint32_t)(uintptr_t)&As[0][0];
  const uint32_t ws_base = (uint32_t)(uintptr_t)&Ws[0][0];
  const uint32_t as_sz   = 128 * A_STRIDE * 4;
  const uint32_t ws_sz   = KC * W_STRIDE * 4;

  // Cooperative async copy of chunk c into buffer buf: 12 instrs per wave.
  auto issue_chunk = [&](int c, int buf) {
    const int k0 = c * KC;
    // A slab: 128 rows x 32 floats -> 1024 x 16B transfers, 8 per thread.
    #pragma unroll
    for (int i = 0; i < 8; ++i) {
      const int q  = i * 128 + t;
      const int r  = q >> 3;
      const int cc = (q & 7) * 4;
      async_ld_b128(as_base + (uint32_t)buf * as_sz + (r * A_STRIDE + cc) * 4,
                    A + (size_t)(m0wg + r) * K_DIM + k0 + cc);
    }
    // W slab: 32 rows x 64 floats -> 512 x 16B transfers, 4 per thread.
    #pragma unroll
    for (int i = 0; i < 4; ++i) {
      const int q  = i * 128 + t;
      const int r  = q >> 4;
      const int cc = (q & 15) * 4;
      async_ld_b128(ws_base + (uint32_t)buf * ws_sz + (r * W_STRIDE + cc) * 4,
                    W + (size_t)(k0 + r) * N_DIM + n0 + cc);
    }
  };

  v8f acc[2][4];
#pragma unroll
  for (int i = 0; i < 2; ++i)
#pragma unroll
    for (int j = 0; j < 4; ++j) acc[i][j] = (v8f)0.0f;

  issue_chunk(0, 0);

  for (int c = 0; c < NC; ++c) {
    const int cur = c & 1;
    if (c + 1 < NC) {
      issue_chunk(c + 1, cur ^ 1);
      asm volatile("s_wait_asynccnt 0xc" ::: "memory"); // chunk c landed
    } else {
      asm volatile("s_wait_asynccnt 0x0" ::: "memory");
    }
    __syncthreads();   // all waves' slab writes visible

    const float* Ab = &As[cur][(wave * 32 + row) * A_STRIDE + kgrp];
    const float* Wb = &Ws[cur][kgrp * W_STRIDE + row];
#pragma unroll
    for (int ks = 0; ks < KC / 4; ++ks) {
      // A frags: 16x4 fp32, lane layout per ISA (VGPR j = A[m][kgrp+j])
      v2f a0 = *(const v2f*)(Ab + ks * 4);
      v2f a1 = *(const v2f*)(Ab + 16 * A_STRIDE + ks * 4);
#pragma unroll
      for (int j = 0; j < 4; ++j) {
        // B frag: 4x16 fp32, VGPR j = W[kgrp+j][n]
        v2f bv;
        bv.x = Wb[ks * 4 * W_STRIDE + j * 16];
        bv.y = Wb[(ks * 4 + 1) * W_STRIDE + j * 16];
        acc[0][j] = __builtin_amdgcn_wmma_f32_16x16x4_f32(
            false, a0, false, bv, (short)0, acc[0][j], false, false);
        acc[1][j] = __builtin_amdgcn_wmma_f32_16x16x4_f32(
            false, a1, false, bv, (short)0, acc[1][j], false, false);
      }
    }
    __syncthreads();   // done reading buffer before it is overwritten
  }

  // Epilogue. C/D layout: VGPR v -> row (v + 8*(lane>>4)), col = lane&15.
  const int mloc = (lane >> 4) * 8;
#pragma unroll
  for (int i = 0; i < 2; ++i) {
#pragma unroll
    for (int j = 0; j < 4; ++j) {
#pragma unroll
      for (int v = 0; v < 8; ++v) {
        const int m = m0 + i * 16 + mloc + v;
        const int n = n0 + j * 16 + row;
        const size_t idx = (size_t)m * N_DIM + n;
        const float val = acc[i][j][v];
        if (mode == 0) {
          out[idx] = val + addv[n];
        } else {
          out[idx] = fast_tanh(val + addv[idx]) - zin[idx];
        }
      }
    }
  }
}

// ---------------------------------------------------------------------------
// Per-row Anderson statistics: num = sum((g-gp)*g), den = sum((g-gp)^2)
// ---------------------------------------------------------------------------
__global__ __launch_bounds__(256) void rowred_kernel(
    const float* __restrict__ g, const float* __restrict__ gp,
    float* __restrict__ nd)
{
  const int r = blockIdx.x;
  __shared__ float sn[256];
  __shared__ float sd[256];
  float num = 0.f, den = 0.f;
  const size_t base = (size_t)r * N_DIM;
  for (int c = threadIdx.x; c < N_DIM; c += 256) {
    const float gv = g[base + c];
    const float d  = gv - gp[base + c];
    num += d * gv;
    den += d * d;
  }
  sn[threadIdx.x] = num;
  sd[threadIdx.x] = den;
  __syncthreads();
  for (int s = 128; s > 0; s >>= 1) {
    if (threadIdx.x < s) {
      sn[threadIdx.x] += sn[threadIdx.x + s];
      sd[threadIdx.x] += sd[threadIdx.x + s];
    }
    __syncthreads();
  }
  if (threadIdx.x == 0) {
    nd[2 * r]     = sn[0];
    nd[2 * r + 1] = sd[0];
  }
}

// ---------------------------------------------------------------------------
// z update. anderson=0: z+BETA*g; anderson=1: z+BETA*g - gamma*(dz+BETA*dg)
// ---------------------------------------------------------------------------
__global__ __launch_bounds__(256) void update_kernel(
    const float* __restrict__ z, const float* __restrict__ g,
    const float* __restrict__ zp, const float* __restrict__ gp,
    const float* __restrict__ nd, float* __restrict__ znew, int anderson)
{
  const size_t idx = (size_t)blockIdx.x * 256 + threadIdx.x;
  const int r = (int)(idx >> 11);           // N_DIM == 2048
  const float zv = z[idx];
  const float gv = g[idx];
  float outv = zv + BETA_C * gv;
  if (anderson) {
    const float gamma = nd[2 * r] / (nd[2 * r + 1] + LAM_C);
    outv -= gamma * ((zv - zp[idx]) + BETA_C * (gv - gp[idx]));
  }
  znew[idx] = outv;
}

__global__ __launch_bounds__(256) void zero_kernel(float* __restrict__ p) {
  p[(size_t)blockIdx.x * 256 + threadIdx.x] = 0.0f;
}

// ---------------------------------------------------------------------------
extern "C" void kernel_launch(void* const* d_in, const int* in_sizes, int n_in,
                              void* d_out, int out_size, void* d_ws, size_t ws_size,
                              hipStream_t stream)
{
  const float* x  = (const float*)d_in[0];   // 1024 x 2048
  const float* Wz = (const float*)d_in[1];   // 2048 x 2048
  const float* Wx = (const float*)d_in[2];   // 2048 x 2048
  const float* b  = (const float*)d_in[3];   // 2048
  float* out = (float*)d_out;                // 1024 x 2048

  const size_t MN = (size_t)M_DIM * N_DIM;
  float* ws    = (float*)d_ws;
  float* xproj = ws;
  float* zb0   = ws + 1 * MN;
  float* zb1   = ws + 2 * MN;
  float* zb2   = ws + 3 * MN;
  float* gb0   = ws + 4 * MN;
  float* gb1   = ws + 5 * MN;
  float* nd    = ws + 6 * MN;                // 2*1024 floats

  const int elem_blocks = (int)(MN / 256);                   // 8192
  const int gemm_blocks = (M_DIM / 128) * (N_DIM / 64);      // 8*32 = 256

  // z_0 = 0 (ws is not re-poisoned between replays; init every call)
  zero_kernel<<<elem_blocks, 256, 0, stream>>>(zb0);

  // x_proj = x @ Wx + b
  gemm_wmma_f32<<<gemm_blocks, 128, 0, stream>>>(x, Wx, b, nullptr, xproj, 0);

  float* zc = zb0; float* zp = zb1; float* zn = zb2;
  float* gc = gb0; float* gp = gb1;

  for (int i = 0; i < 6; ++i) {
    // g = tanh(z @ Wz + x_proj) - z
    gemm_wmma_f32<<<gemm_blocks, 128, 0, stream>>>(zc, Wz, xproj, zc, gc, 1);

    const int anderson = (i >= 2) ? 1 : 0;
    if (anderson) {
      rowred_kernel<<<M_DIM, 256, 0, stream>>>(gc, gp, nd);
    }
    float* dst = (i == 5) ? out : zn;
    update_kernel<<<elem_blocks, 256, 0, stream>>>(zc, gc, zp, gp, nd, dst, anderson);

    // rotate: (zp, zc, zn) <- (zc, zn, zp) ; (gp, gc) <- (gc, gp)
    float* tz = zp; zp = zc; zc = zn; zn = tz;
    float* tg = gp; gp = gc; gc = tg;
  }
}